// SS2D_81647328296973
// MI455X (gfx1250) — compile-verified
//
#include <hip/hip_runtime.h>
#include <math.h>

// ---------------- problem constants ----------------
#define BB    8
#define LL    4096      // H*W
#define NROW  32768     // B*L
#define CIN   96
#define DI    192
#define D2    384
#define SP    96
#define DSN   8
#define DTR   6
#define CXP_C 22
#define NCHNK 64
#define CHLEN 64

typedef float v2f __attribute__((ext_vector_type(2)));
typedef float v8f __attribute__((ext_vector_type(8)));

__device__ __forceinline__ v8f wmma4(v2f a, v2f b, v8f c) {
  // V_WMMA_F32_16X16X4_F32 : D = A(16x4) * B(4x16) + C(16x16), all fp32
  return __builtin_amdgcn_wmma_f32_16x16x4_f32(false, a, false, b, (short)0, c,
                                               false, false);
}

// store one 16x16 C/D tile; mbase = first M row of this half-wave's 8 rows
__device__ __forceinline__ void store_tile(float* __restrict__ Out, const v8f& c,
                                           int mbase, int n, bool nok,
                                           int N, int out_mode, int NC) {
  if (!nok) return;
  if (out_mode == 0) {
#pragma unroll
    for (int r = 0; r < 8; ++r) Out[(size_t)(mbase + r) * N + n] = c[r];
  } else {
#pragma unroll
    for (int r = 0; r < 8; ++r) {
      int row = mbase + r;
      int b = row >> 12, l = row & 4095;
      Out[((size_t)b * NC + n) * LL + l] = c[r];
    }
  }
}

// ------- GEMM1: xz = NCHW(x) @ W_in^T -> [NROW][384], 32x32 block / wave -------
__global__ void k_gemm_in(const float* __restrict__ X,
                          const float* __restrict__ Wt,
                          float* __restrict__ Out) {
  const int tilesN2 = D2 / 32;                       // 12
  int wid  = blockIdx.x * (blockDim.x >> 5) + (threadIdx.x >> 5);
  int lane = threadIdx.x & 31;
  int tm = wid / tilesN2, tn = wid - tm * tilesN2;
  int lh = lane & 15, hi = lane >> 4, kh = hi << 1;
  int m0 = tm * 32 + lh, m1 = m0 + 16;               // same b (4096 % 32 == 0)
  int n0 = tn * 32 + lh, n1 = n0 + 16;
  int b = m0 >> 12;
  const float* xb0 = X + (size_t)b * CIN * LL + (m0 & 4095);  // stride LL in k
  const float* xb1 = X + (size_t)b * CIN * LL + (m1 & 4095);
  const float* wr0 = Wt + (size_t)n0 * CIN;
  const float* wr1 = Wt + (size_t)n1 * CIN;
  v8f c00, c01, c10, c11;
#pragma unroll
  for (int r = 0; r < 8; ++r) { c00[r] = 0.f; c01[r] = 0.f; c10[r] = 0.f; c11[r] = 0.f; }
  for (int k = 0; k < CIN; k += 4) {
    int k0 = k + kh;
    v2f a0, a1, b0, b1;
    a0.x = xb0[(size_t)k0 * LL];       a0.y = xb0[(size_t)(k0 + 1) * LL];
    a1.x = xb1[(size_t)k0 * LL];       a1.y = xb1[(size_t)(k0 + 1) * LL];
    b0.x = wr0[k0];                    b0.y = wr0[k0 + 1];
    b1.x = wr1[k0];                    b1.y = wr1[k0 + 1];
    c00 = wmma4(a0, b0, c00);
    c01 = wmma4(a0, b1, c01);
    c10 = wmma4(a1, b0, c10);
    c11 = wmma4(a1, b1, c11);
  }
  int mb0 = tm * 32 + hi * 8, mb1 = mb0 + 16;
  store_tile(Out, c00, mb0, n0, true, D2, 0, 0);
  store_tile(Out, c01, mb0, n1, true, D2, 0, 0);
  store_tile(Out, c10, mb1, n0, true, D2, 0, 0);
  store_tile(Out, c11, mb1, n1, true, D2, 0, 0);
}

// ------------- generic GEMM: Out = A[:,aoff:aoff+K] @ W^T (+bias) -------------
// 32x32 block per wave (2x2 of 16x16 WMMA tiles).
// A row-major [NROW][lda]; W row-major [N][K]; out_mode 0: Out[row][N] (ld=N)
// out_mode 1: NCHW scatter Out[(b*NC+n)*LL + l], row = b*4096 + l
__global__ void k_gemm_rm(const float* __restrict__ A, int lda, int aoff,
                          const float* __restrict__ Wt,
                          const float* __restrict__ bias,
                          float* __restrict__ Out,
                          int K, int N, int tilesN2, int out_mode, int NC) {
  int wid  = blockIdx.x * (blockDim.x >> 5) + (threadIdx.x >> 5);
  int lane = threadIdx.x & 31;
  int tm = wid / tilesN2, tn = wid - tm * tilesN2;
  int lh = lane & 15, hi = lane >> 4, kh = hi << 1;
  int m0 = tm * 32 + lh, m1 = m0 + 16;
  int n0 = tn * 32 + lh, n1 = n0 + 16;
  bool nok0 = (n0 < N), nok1 = (n1 < N);
  const float* ar0 = A + (size_t)m0 * lda + aoff;
  const float* ar1 = A + (size_t)m1 * lda + aoff;
  const float* wr0 = Wt + (size_t)(nok0 ? n0 : 0) * K;  // safe row for masked lanes
  const float* wr1 = Wt + (size_t)(nok1 ? n1 : 0) * K;
  v8f c00, c01, c10, c11;
  float bv0 = (bias != nullptr && nok0) ? bias[n0] : 0.0f;
  float bv1 = (bias != nullptr && nok1) ? bias[n1] : 0.0f;
#pragma unroll
  for (int r = 0; r < 8; ++r) { c00[r] = bv0; c01[r] = bv1; c10[r] = bv0; c11[r] = bv1; }
  for (int k = 0; k < K; k += 4) {
    int k0 = k + kh;
    v2f a0, a1, b0, b1;
    a0.x = ar0[k0];  a0.y = ar0[k0 + 1];
    a1.x = ar1[k0];  a1.y = ar1[k0 + 1];
    b0.x = wr0[k0];  b0.y = wr0[k0 + 1];
    b1.x = wr1[k0];  b1.y = wr1[k0 + 1];
    c00 = wmma4(a0, b0, c00);        // unconditional: EXEC stays all-ones
    c01 = wmma4(a0, b1, c01);
    c10 = wmma4(a1, b0, c10);
    c11 = wmma4(a1, b1, c11);
  }
  int mb0 = tm * 32 + hi * 8, mb1 = mb0 + 16;
  store_tile(Out, c00, mb0, n0, nok0, N, out_mode, NC);
  store_tile(Out, c01, mb0, n1, nok1, N, out_mode, NC);
  store_tile(Out, c10, mb1, n0, nok0, N, out_mode, NC);
  store_tile(Out, c11, mb1, n1, nok1, N, out_mode, NC);
}

// ---------- depthwise mix: id | dw3 | dw5 | dw7 on t[384], then exact GELU ----------
__global__ void k_dwmix(const float* __restrict__ T,
                        const float* __restrict__ w3, const float* __restrict__ b3,
                        const float* __restrict__ w5, const float* __restrict__ b5,
                        const float* __restrict__ w7, const float* __restrict__ b7,
                        float* __restrict__ O) {
  int idx = blockIdx.x * blockDim.x + threadIdx.x;
  if (idx >= NROW * D2) return;
  int c = idx % D2;
  int row = idx / D2;
  int b = row >> 12, h = (row >> 6) & 63, w = row & 63;
  float v;
  if (c < SP) {
    v = T[(size_t)row * D2 + c];
  } else {
    int ksz, ci;
    const float *wp, *bp;
    if (c < 2 * SP)      { ksz = 3; wp = w3; bp = b3; ci = c - SP; }
    else if (c < 3 * SP) { ksz = 5; wp = w5; bp = b5; ci = c - 2 * SP; }
    else                 { ksz = 7; wp = w7; bp = b7; ci = c - 3 * SP; }
    int pad = ksz >> 1;
    float acc = bp[ci];
    for (int kh = 0; kh < ksz; ++kh) {
      int hy = h + kh - pad;
      if (hy < 0 || hy >= 64) continue;
      for (int kw = 0; kw < ksz; ++kw) {
        int wx = w + kw - pad;
        if (wx < 0 || wx >= 64) continue;
        acc += wp[ci * ksz * ksz + kh * ksz + kw] *
               T[((size_t)(b * 4096 + hy * 64 + wx)) * D2 + c];
      }
    }
    v = acc;
  }
  O[(size_t)row * D2 + c] = 0.5f * v * (1.0f + erff(v * 0.7071067811865475f));
}

// ---------- xc = silu(dw3x3(xm) + b), xm = xz channels [0,192) ----------
__global__ void k_dwsilu(const float* __restrict__ XZ,
                         const float* __restrict__ wc, const float* __restrict__ bc,
                         float* __restrict__ XC) {
  int idx = blockIdx.x * blockDim.x + threadIdx.x;
  if (idx >= NROW * DI) return;
  int c = idx % DI;
  int row = idx / DI;
  int b = row >> 12, h = (row >> 6) & 63, w = row & 63;
  float acc = bc[c];
  for (int kh = 0; kh < 3; ++kh) {
    int hy = h + kh - 1;
    if (hy < 0 || hy >= 64) continue;
    for (int kw = 0; kw < 3; ++kw) {
      int wx = w + kw - 1;
      if (wx < 0 || wx >= 64) continue;
      acc += wc[c * 9 + kh * 3 + kw] *
             XZ[((size_t)(b * 4096 + hy * 64 + wx)) * D2 + c];
    }
  }
  XC[(size_t)row * DI + c] = acc / (1.0f + expf(-acc));
}

// ---------- haar 2x2 subsample combine -> yL, y_mean  [b][8][4096] ----------
__global__ void k_haarpost(const float* __restrict__ XW,
                           float* __restrict__ YL, float* __restrict__ YM) {
  int idx = blockIdx.x * blockDim.x + threadIdx.x;   // 8*32*32*32
  if (idx >= 262144) return;
  int j = idx & 31, i = (idx >> 5) & 31, ch = (idx >> 10) & 31, b = idx >> 15;
  size_t r00 = ((size_t)(b * 4096 + (2 * i) * 64 + 2 * j)) * 32 + ch;
  float a  = XW[r00];
  float b2 = XW[r00 + 32];                 // (2i, 2j+1)
  float c2 = XW[r00 + 64 * 32];            // (2i+1, 2j)
  float d2 = XW[r00 + 65 * 32];            // (2i+1, 2j+1)
  int d = ch >> 2;
  int l = ((ch & 3) << 10) + i * 32 + j;
  size_t o = ((size_t)b * DSN + d) * LL + l;
  YL[o] = 0.5f * (a + b2 + c2 + d2);
  YM[o] = (3.0f * a - b2 - c2 - d2) * (1.0f / 6.0f);
}

// ---------- depthwise conv1d (k=7, pad 3) along L on x_dbl [b][22][L] ----------
__global__ void k_conv1d(const float* __restrict__ XD,
                         const float* __restrict__ Wx, const float* __restrict__ bx,
                         float* __restrict__ XO) {
  int idx = blockIdx.x * blockDim.x + threadIdx.x;   // 8*22*4096
  if (idx >= BB * CXP_C * LL) return;
  int l = idx & 4095;
  int bc = idx >> 12;
  int c = bc % CXP_C;
  float acc = bx[c];
  const float* src = XD + (size_t)bc * LL;
  for (int t = 0; t < 7; ++t) {
    int ll2 = l + t - 3;
    if (ll2 >= 0 && ll2 < LL) acc += Wx[c * 7 + t] * src[ll2];
  }
  XO[idx] = acc;
}

// ---------- delta = softplus(dtw @ dts + dt_bias)  [bd][L] ----------
__global__ void k_delta(const float* __restrict__ XD2,
                        const float* __restrict__ dtw, const float* __restrict__ dtb,
                        float* __restrict__ DL) {
  int idx = blockIdx.x * blockDim.x + threadIdx.x;   // 8*192*4096
  if (idx >= BB * DI * LL) return;
  int l = idx & 4095;
  int bd = idx >> 12;
  int b = bd / DI, d = bd - b * DI;
  float raw = dtb[d];
#pragma unroll
  for (int r = 0; r < DTR; ++r)
    raw += dtw[d * DTR + r] * XD2[((size_t)b * CXP_C + r) * LL + l];
  DL[idx] = (raw > 20.0f) ? raw : log1pf(expf(raw));
}

// ---------- Bs, Cs from x_dbl * (yL, y_mean) + 1e-6 ----------
__global__ void k_bscs(const float* __restrict__ XD2,
                       const float* __restrict__ YL, const float* __restrict__ YM,
                       float* __restrict__ BS, float* __restrict__ CS) {
  int idx = blockIdx.x * blockDim.x + threadIdx.x;   // 8*8*4096
  if (idx >= BB * DSN * LL) return;
  int l = idx & 4095;
  int bn = idx >> 12;
  int b = bn >> 3;
  BS[idx] = XD2[((size_t)b * CXP_C + DTR + ((bn & 7))) * LL + l] * YL[idx] + 1e-6f;
  CS[idx] = XD2[((size_t)b * CXP_C + DTR + DSN + ((bn & 7))) * LL + l] * YM[idx] + 1e-6f;
}

// ---------- chunked selective scan: pass 1 (local scan + A-products) ----------
__global__ void k_scan1(const float* __restrict__ DL, const float* __restrict__ XC,
                        const float* __restrict__ BS, const float* __restrict__ Alog,
                        float* __restrict__ CHA, float* __restrict__ CHH) {
  int t = blockIdx.x * blockDim.x + threadIdx.x;     // (b*192+d)*64 + chunk
  if (t >= BB * DI * NCHNK) return;
  int bd = t >> 6, ch = t & 63;
  int b = bd / DI, d = bd - b * DI;
  float Ad[DSN], h[DSN], ap[DSN];
#pragma unroll
  for (int n = 0; n < DSN; ++n) { Ad[n] = -expf(Alog[d * DSN + n]); h[n] = 0.0f; ap[n] = 1.0f; }
  int l0 = ch * CHLEN;
  for (int i = 0; i < CHLEN; ++i) {
    int l = l0 + i;
    float dlt = DL[(size_t)bd * LL + l];
    float u   = XC[((size_t)b * 4096 + l) * DI + d];
    float du  = dlt * u;
#pragma unroll
    for (int n = 0; n < DSN; ++n) {
      float dA = expf(dlt * Ad[n]);
      h[n]  = dA * h[n] + du * BS[((size_t)b * DSN + n) * LL + l];
      ap[n] = dA * ap[n];
    }
  }
#pragma unroll
  for (int n = 0; n < DSN; ++n) { CHA[(size_t)t * DSN + n] = ap[n]; CHH[(size_t)t * DSN + n] = h[n]; }
}

// ---------- pass 2: sequential chunk combine (per bd row) ----------
__global__ void k_scan2(const float* __restrict__ CHA, const float* __restrict__ CHH,
                        float* __restrict__ CHI) {
  int bd = blockIdx.x * blockDim.x + threadIdx.x;
  if (bd >= BB * DI) return;
  float hin[DSN];
#pragma unroll
  for (int n = 0; n < DSN; ++n) hin[n] = 0.0f;
  for (int ch = 0; ch < NCHNK; ++ch) {
    size_t base = ((size_t)bd * NCHNK + ch) * DSN;
#pragma unroll
    for (int n = 0; n < DSN; ++n) {
      CHI[base + n] = hin[n];
      hin[n] = CHA[base + n] * hin[n] + CHH[base + n];
    }
  }
}

// ---------- pass 3: re-scan with true h_in, emit y (NHWC) ----------
__global__ void k_scan3(const float* __restrict__ DL, const float* __restrict__ XC,
                        const float* __restrict__ BS, const float* __restrict__ CS,
                        const float* __restrict__ Alog, const float* __restrict__ Dsv,
                        const float* __restrict__ CHI, float* __restrict__ Y) {
  int t = blockIdx.x * blockDim.x + threadIdx.x;
  if (t >= BB * DI * NCHNK) return;
  int bd = t >> 6, ch = t & 63;
  int b = bd / DI, d = bd - b * DI;
  float Ad[DSN], h[DSN];
#pragma unroll
  for (int n = 0; n < DSN; ++n) { Ad[n] = -expf(Alog[d * DSN + n]); h[n] = CHI[(size_t)t * DSN + n]; }
  float Dd = Dsv[d];
  int l0 = ch * CHLEN;
  for (int i = 0; i < CHLEN; ++i) {
    int l = l0 + i;
    float dlt = DL[(size_t)bd * LL + l];
    float u   = XC[((size_t)b * 4096 + l) * DI + d];
    float du  = dlt * u;
    float y = 0.0f;
#pragma unroll
    for (int n = 0; n < DSN; ++n) {
      float dA = expf(dlt * Ad[n]);
      h[n] = dA * h[n] + du * BS[((size_t)b * DSN + n) * LL + l];
      y += h[n] * CS[((size_t)b * DSN + n) * LL + l];
    }
    Y[((size_t)b * 4096 + l) * DI + d] = y + Dd * u;
  }
}

// ---------- LayerNorm(192) * silu(z with (h,w) swap) ; wave per row ----------
__global__ void k_lnmul(const float* __restrict__ Y, const float* __restrict__ ZF,
                        const float* __restrict__ gamma, const float* __restrict__ beta,
                        float* __restrict__ O) {
  int wid  = blockIdx.x * (blockDim.x >> 5) + (threadIdx.x >> 5);   // row
  int lane = threadIdx.x & 31;
  const float* yr = Y + (size_t)wid * DI;
  float vals[6];
  float s = 0.0f, s2 = 0.0f;
#pragma unroll
  for (int i = 0; i < 6; ++i) {
    float v = yr[lane + i * 32];
    vals[i] = v; s += v; s2 += v * v;
  }
#pragma unroll
  for (int o = 16; o > 0; o >>= 1) {
    s  += __shfl_xor(s, o, 32);
    s2 += __shfl_xor(s2, o, 32);
  }
  float mu  = s * (1.0f / DI);
  float var = s2 * (1.0f / DI) - mu * mu;
  float inv = rsqrtf(var + 1e-5f);
  int b = wid >> 12, h = (wid >> 6) & 63, w = wid & 63;
  const float* zr = ZF + ((size_t)(b * 4096 + w * 64 + h)) * DI;   // (0,3,2,1) transpose folded in
  float* orow = O + (size_t)wid * DI;
#pragma unroll
  for (int i = 0; i < 6; ++i) {
    int ci = lane + i * 32;
    float zv = zr[ci];
    float sz = zv / (1.0f + expf(-zv));
    orow[ci] = ((vals[i] - mu) * inv * gamma[ci] + beta[ci]) * sz;
  }
}

// ------------------------------- launcher -------------------------------
extern "C" void kernel_launch(void* const* d_in, const int* in_sizes, int n_in,
                              void* d_out, int out_size, void* d_ws, size_t ws_size,
                              hipStream_t stream) {
  (void)in_sizes; (void)n_in; (void)out_size; (void)ws_size;
  const float* x          = (const float*)d_in[0];
  const float* W_in       = (const float*)d_in[1];
  const float* W_dwc_init = (const float*)d_in[2];
  const float* b_dwc_init = (const float*)d_in[3];
  const float* W_dw3      = (const float*)d_in[4];
  const float* b_dw3      = (const float*)d_in[5];
  const float* W_dw5      = (const float*)d_in[6];
  const float* b_dw5      = (const float*)d_in[7];
  const float* W_dw7      = (const float*)d_in[8];
  const float* b_dw7      = (const float*)d_in[9];
  const float* W_dwc_fin  = (const float*)d_in[10];
  const float* b_dwc_fin  = (const float*)d_in[11];
  const float* W_conv2d   = (const float*)d_in[12];
  const float* b_conv2d   = (const float*)d_in[13];
  const float* W_haar     = (const float*)d_in[14];
  const float* x_proj_w   = (const float*)d_in[15];
  const float* W_xconv    = (const float*)d_in[16];
  const float* b_xconv    = (const float*)d_in[17];
  const float* dt_projs_w = (const float*)d_in[18];
  const float* dt_projs_b = (const float*)d_in[19];
  const float* A_logs     = (const float*)d_in[20];
  const float* Ds         = (const float*)d_in[21];
  const float* ln_gamma   = (const float*)d_in[22];
  const float* ln_beta    = (const float*)d_in[23];
  const float* W_out      = (const float*)d_in[24];
  float* out = (float*)d_out;

  float* ws = (float*)d_ws;
  // big buffers
  float* XZ   = ws;                        // [NROW][384]   12,582,912 f
  float* T1   = ws + 12582912;             // [NROW][384]   (t ; later y_nhwc)
  float* T2   = ws + 25165824;             // [NROW][384]   (t2; later y_ln)
  float* ZFIN = ws + 37748736;             // [NROW][192]
  float* XC   = ws + 44040192;             // [NROW][192]
  // small buffers overlap XZ (dead after k_dwsilu)
  float* XW    = XZ;                       // [NROW][32]
  float* YL    = XZ + 1048576;             // [8][8][4096]
  float* YM    = XZ + 1310720;
  float* XDBL  = XZ + 1572864;             // [8][22][4096]
  float* XDB2  = XZ + 2293760;
  float* DELTA = XZ + 3014656;             // [8*192][4096]
  float* BS    = XZ + 9306112;             // [8][8][4096]
  float* CS    = XZ + 9568256;
  float* CHA   = XZ + 9830400;             // [1536*64][8]
  float* CHH   = XZ + 10616832;
  float* CHI   = XZ + 11403264;
  float* YN    = T1;                       // y (NHWC)
  float* YLN   = T2;                       // LN(y)*silu(z)

  // 32x32 blocks: tilesM2 = NROW/32 = 1024; waves = 1024 * tilesN2; blocks = waves/8
  // 1) xz = x(NCHW) @ W_in^T                       [32768 x 96 x 384]
  k_gemm_in<<<1536, 256, 0, stream>>>(x, W_in, XZ);
  // 2) t = z @ W_dwc_init^T + b                    [32768 x 192 x 384]
  k_gemm_rm<<<1536, 256, 0, stream>>>(XZ, D2, DI, W_dwc_init, b_dwc_init, T1,
                                      DI, D2, 12, 0, 0);
  // 3) depthwise mix (id|3x3|5x5|7x7) + GELU
  k_dwmix<<<49152, 256, 0, stream>>>(T1, W_dw3, b_dw3, W_dw5, b_dw5, W_dw7, b_dw7, T2);
  // 4) zfin = gelu_t @ W_dwc_fin^T + b             [32768 x 384 x 192]
  k_gemm_rm<<<768, 256, 0, stream>>>(T2, D2, 0, W_dwc_fin, b_dwc_fin, ZFIN,
                                     D2, DI, 6, 0, 0);
  // 5) xc = silu(dw3x3(xm) + b)
  k_dwsilu<<<24576, 256, 0, stream>>>(XZ, W_conv2d, b_conv2d, XC);
  // 6) xw = xc @ W_haar^T                          [32768 x 192 x 32]
  k_gemm_rm<<<128, 256, 0, stream>>>(XC, DI, 0, W_haar, nullptr, XW,
                                     DI, 32, 1, 0, 0);
  // 7) yL / y_mean from 2x2 subsampling
  k_haarpost<<<1024, 256, 0, stream>>>(XW, YL, YM);
  // 8) x_dbl = xc @ x_proj^T  -> [b][22][L]        [32768 x 192 x 22]
  k_gemm_rm<<<128, 256, 0, stream>>>(XC, DI, 0, x_proj_w, nullptr, XDBL,
                                     DI, CXP_C, 1, 1, CXP_C);
  // 9) depthwise conv1d k=7 along L
  k_conv1d<<<2816, 256, 0, stream>>>(XDBL, W_xconv, b_xconv, XDB2);
  // 10) delta = softplus(dtw @ dts + dt_bias)
  k_delta<<<24576, 256, 0, stream>>>(XDB2, dt_projs_w, dt_projs_b, DELTA);
  // 11) Bs, Cs
  k_bscs<<<1024, 256, 0, stream>>>(XDB2, YL, YM, BS, CS);
  // 12-14) chunked selective scan -> y (NHWC)
  k_scan1<<<384, 256, 0, stream>>>(DELTA, XC, BS, A_logs, CHA, CHH);
  k_scan2<<<6, 256, 0, stream>>>(CHA, CHH, CHI);
  k_scan3<<<384, 256, 0, stream>>>(DELTA, XC, BS, CS, A_logs, Ds, CHI, YN);
  // 15) LayerNorm * silu(z^T)
  k_lnmul<<<4096, 256, 0, stream>>>(YN, ZFIN, ln_gamma, ln_beta, YLN);
  // 16) out = y_ln @ W_out^T, scattered to NCHW    [32768 x 192 x 96]
  k_gemm_rm<<<384, 256, 0, stream>>>(YLN, DI, 0, W_out, nullptr, out,
                                     DI, CIN, 3, 1, CIN);
}